// GIN_45208825757773
// MI455X (gfx1250) — compile-verified
//
#include <hip/hip_runtime.h>
#include <hip/hip_bf16.h>

typedef __attribute__((ext_vector_type(2))) float v2f;
typedef __attribute__((ext_vector_type(8))) float v8f;

#define N_NODES   100000
#define N_EDGES   1600000
#define NUM_GRAPHS 64

// ---------------------------------------------------------------------------
// Elementwise helpers
// ---------------------------------------------------------------------------
__global__ void copy_f4_kernel(const float4* __restrict__ in, float4* __restrict__ out,
                               long long n4) {
    long long i = (long long)blockIdx.x * blockDim.x + threadIdx.x;
    if (i < n4) out[i] = in[i];
}

__global__ void zero_kernel(float* __restrict__ p, long long n) {
    long long i = (long long)blockIdx.x * blockDim.x + threadIdx.x;
    if (i < n) p[i] = 0.0f;
}

// ---------------------------------------------------------------------------
// Edge scatter-add:  out[dst[e]] += feat[src[e]]   (out pre-initialized = feat)
// CH/4 threads per edge, float4 gather, 4x global_atomic_add_f32 scatter.
// ---------------------------------------------------------------------------
template <int CH>
__global__ void scatter_add_kernel(const float* __restrict__ feat,
                                   const long long* __restrict__ ei,
                                   float* __restrict__ out) {
    constexpr int TPE = CH / 4;
    long long tid = (long long)blockIdx.x * blockDim.x + threadIdx.x;
    long long e = tid / TPE;
    int c4 = (int)(tid % TPE);
    if (e >= (long long)N_EDGES) return;
    long long s = ei[e];
    long long d = ei[(long long)N_EDGES + e];
    const float4 v = *(const float4*)(feat + s * CH + c4 * 4);
    float* o = out + d * CH + c4 * 4;
    atomicAdd(o + 0, v.x);
    atomicAdd(o + 1, v.y);
    atomicAdd(o + 2, v.z);
    atomicAdd(o + 3, v.w);
}

// ---------------------------------------------------------------------------
// GEMM: out[M x N] = act(in[M x K] @ W[K x N] + b), f32 WMMA 16x16x4.
// 256 threads = 8 waves; each wave computes a 16 x N strip.
// W staged transposed in LDS with +1 pad to avoid bank conflicts.
// ---------------------------------------------------------------------------
template <int K, int N, bool RELU>
__global__ __launch_bounds__(256) void gemm_wmma_kernel(
    const float* __restrict__ in, const float* __restrict__ W,
    const float* __restrict__ bias, float* __restrict__ out, int M) {
    constexpr int NT = N / 16;
    constexpr int KP = K + 1;            // padded LDS row (bank-conflict free)
    __shared__ float wlds[N * KP];
    __shared__ float blds[N];

    const int tid = threadIdx.x;
    for (int i = tid; i < K * N; i += 256) {
        int k = i / N, n = i % N;
        wlds[n * KP + k] = W[i];         // transpose: wlds[n][k]
    }
    for (int i = tid; i < N; i += 256) blds[i] = bias[i];
    __syncthreads();

    const int wave = tid >> 5;
    const int lane = tid & 31;
    const int mbase = (blockIdx.x * 8 + wave) * 16;
    if (mbase >= M) return;              // whole-wave uniform exit: EXEC stays all-1

    const int mrow = mbase + (lane & 15);
    const int koff = (lane >> 4) * 2;    // lanes 0-15: K{0,1}; lanes 16-31: K{2,3}
    const float* arow = in + (long long)mrow * K + koff;

    v8f acc[NT] = {};
    for (int k = 0; k < K; k += 4) {
        v2f a;
        a[0] = arow[k];
        a[1] = arow[k + 1];
#pragma unroll
        for (int nt = 0; nt < NT; ++nt) {
            const int col = nt * 16 + (lane & 15);
            v2f b;
            b[0] = wlds[col * KP + k + koff];
            b[1] = wlds[col * KP + k + koff + 1];
            acc[nt] = __builtin_amdgcn_wmma_f32_16x16x4_f32(
                false, a, false, b, (short)0, acc[nt], false, false);
        }
    }

    const int colbase = lane & 15;
    const int rshift = (lane >> 4) * 8;  // C/D layout: vgpr i -> M=i (lo) / M=i+8 (hi)
#pragma unroll
    for (int nt = 0; nt < NT; ++nt) {
        const int col = nt * 16 + colbase;
        const float bv = blds[col];
#pragma unroll
        for (int i = 0; i < 8; ++i) {
            const int row = mbase + i + rshift;
            float v = acc[nt][i] + bv;
            if (RELU) v = fmaxf(v, 0.0f);
            out[(long long)row * N + col] = v;
        }
    }
}

// ---------------------------------------------------------------------------
// Segment-sum pool over sorted batch ids (atomic; 64x128 target lives in L2)
// ---------------------------------------------------------------------------
__global__ void pool_sum_kernel(const float* __restrict__ h,
                                const long long* __restrict__ batch,
                                float* __restrict__ pooled) {
    long long tid = (long long)blockIdx.x * blockDim.x + threadIdx.x;
    int node = (int)(tid >> 5);          // 32 threads per node (128 ch / 4)
    int c4 = (int)(tid & 31);
    if (node >= N_NODES) return;
    int g = (int)batch[node];
    const float4 v = *(const float4*)(h + (long long)node * 128 + c4 * 4);
    float* o = pooled + g * 128 + c4 * 4;
    atomicAdd(o + 0, v.x);
    atomicAdd(o + 1, v.y);
    atomicAdd(o + 2, v.z);
    atomicAdd(o + 3, v.w);
}

__global__ void count_kernel(const long long* __restrict__ batch,
                             float* __restrict__ counts) {
    int n = blockIdx.x * blockDim.x + threadIdx.x;
    if (n >= N_NODES) return;
    atomicAdd(&counts[(int)batch[n]], 1.0f);
}

// ---------------------------------------------------------------------------
// Head: mean-pool divide + 128x10 FC + log_softmax. One thread per graph.
// ---------------------------------------------------------------------------
__global__ void head_kernel(const float* __restrict__ pooled,
                            const float* __restrict__ counts,
                            const float* __restrict__ Wfc,
                            const float* __restrict__ bfc,
                            float* __restrict__ out) {
    int g = threadIdx.x;
    if (g >= NUM_GRAPHS) return;
    float inv = 1.0f / fmaxf(counts[g], 1.0f);
    float logits[10];
#pragma unroll
    for (int o = 0; o < 10; ++o) {
        float acc = bfc[o];
        for (int c = 0; c < 128; ++c)
            acc += pooled[g * 128 + c] * inv * Wfc[c * 10 + o];
        logits[o] = acc;
    }
    float m = logits[0];
#pragma unroll
    for (int o = 1; o < 10; ++o) m = fmaxf(m, logits[o]);
    float s = 0.0f;
#pragma unroll
    for (int o = 0; o < 10; ++o) s += expf(logits[o] - m);
    float lse = m + logf(s);
#pragma unroll
    for (int o = 0; o < 10; ++o) out[g * 10 + o] = logits[o] - lse;
}

// ---------------------------------------------------------------------------
extern "C" void kernel_launch(void* const* d_in, const int* in_sizes, int n_in,
                              void* d_out, int out_size, void* d_ws, size_t ws_size,
                              hipStream_t stream) {
    const float* x = (const float*)d_in[0];
    const long long* ei = (const long long*)d_in[1];   // int64 [2, E]
    const long long* batch = (const long long*)d_in[2];// int64 [N]
    const float* W1 = (const float*)d_in[3];
    const float* b1 = (const float*)d_in[4];
    const float* W2 = (const float*)d_in[5];
    const float* b2 = (const float*)d_in[6];
    const float* W3 = (const float*)d_in[7];
    const float* b3 = (const float*)d_in[8];
    const float* W4 = (const float*)d_in[9];
    const float* b4 = (const float*)d_in[10];
    const float* Wfc = (const float*)d_in[11];
    const float* bfc = (const float*)d_in[12];
    float* out = (float*)d_out;

    // Workspace layout (floats)
    float* A = (float*)d_ws;                       // N_NODES x 128
    float* B = A + (long long)N_NODES * 128;       // N_NODES x 64
    float* C = B + (long long)N_NODES * 64;        // N_NODES x 64
    float* pooled = C + (long long)N_NODES * 64;   // 64 x 128
    float* counts = pooled + NUM_GRAPHS * 128;     // 64

    const int BS = 256;
    const int gemm_blocks = ((N_NODES / 16) + 7) / 8;   // 782

    // ---- agg1: A = x + scatter(x[src] -> dst), 128 channels ----
    {
        long long n4 = (long long)N_NODES * 128 / 4;
        copy_f4_kernel<<<(int)((n4 + BS - 1) / BS), BS, 0, stream>>>(
            (const float4*)x, (float4*)A, n4);
        long long th = (long long)N_EDGES * 32;
        scatter_add_kernel<128><<<(int)((th + BS - 1) / BS), BS, 0, stream>>>(x, ei, A);
    }
    // ---- MLP1: B = relu(A@W1+b1); C = relu(B@W2+b2) ----
    gemm_wmma_kernel<128, 64, true><<<gemm_blocks, BS, 0, stream>>>(A, W1, b1, B, N_NODES);
    gemm_wmma_kernel<64, 64, true><<<gemm_blocks, BS, 0, stream>>>(B, W2, b2, C, N_NODES);

    // ---- agg2: B = C + scatter(C[src] -> dst), 64 channels ----
    {
        long long n4 = (long long)N_NODES * 64 / 4;
        copy_f4_kernel<<<(int)((n4 + BS - 1) / BS), BS, 0, stream>>>(
            (const float4*)C, (float4*)B, n4);
        long long th = (long long)N_EDGES * 16;
        scatter_add_kernel<64><<<(int)((th + BS - 1) / BS), BS, 0, stream>>>(C, ei, B);
    }
    // ---- MLP2: C = relu(B@W3+b3); A = C@W4+b4 ----
    gemm_wmma_kernel<64, 64, true><<<gemm_blocks, BS, 0, stream>>>(B, W3, b3, C, N_NODES);
    gemm_wmma_kernel<64, 128, false><<<gemm_blocks, BS, 0, stream>>>(C, W4, b4, A, N_NODES);

    // ---- mean pool + head ----
    {
        long long nz = NUM_GRAPHS * 128 + NUM_GRAPHS;
        zero_kernel<<<(int)((nz + BS - 1) / BS), BS, 0, stream>>>(pooled, nz);
        long long th = (long long)N_NODES * 32;
        pool_sum_kernel<<<(int)((th + BS - 1) / BS), BS, 0, stream>>>(A, batch, pooled);
        count_kernel<<<(N_NODES + BS - 1) / BS, BS, 0, stream>>>(batch, counts);
        head_kernel<<<1, 64, 0, stream>>>(pooled, counts, Wfc, bfc, out);
    }
}